// APPNPConv_20942260536144
// MI455X (gfx1250) — compile-verified
//
#include <hip/hip_runtime.h>
#include <cmath>
#include <cstdint>

#define HIDDEN 64
#define KSTEPS 10
#define BETA_W 2.0
#define TILE_EDGES 1024

// ---------------------------------------------------------------------------
// CDNA5 helpers
// ---------------------------------------------------------------------------

// Non-returning fp32 atomic add at device scope -> GLOBAL_ATOMIC_ADD_F32
// (STOREcnt-tracked, executes in the L2 atomic units, no VGPR writeback).
__device__ __forceinline__ void atomic_add_f32_dev(float* p, float v) {
    asm volatile("global_atomic_add_f32 %0, %1, off scope:SCOPE_DEV"
                 :: "v"(p), "v"(v)
                 : "memory");
}

// Per-lane async bulk copy 16B global -> LDS (ASYNCcnt-tracked).
__device__ __forceinline__ void async_load_b128_to_lds(uint32_t lds_off, const void* gaddr) {
    asm volatile("global_load_async_to_lds_b128 %0, %1, off"
                 :: "v"(lds_off), "v"(gaddr)
                 : "memory");
}

__device__ __forceinline__ void wait_asynccnt0() {
    asm volatile("s_wait_asynccnt 0x0" ::: "memory");
}

__device__ __forceinline__ void wait_tensorcnt0() {
#if __has_builtin(__builtin_amdgcn_s_wait_tensorcnt)
    __builtin_amdgcn_s_wait_tensorcnt(0);
#else
    asm volatile("s_wait_tensorcnt 0x0" ::: "memory");
#endif
}

// Low 32 bits of a flat pointer into the LDS aperture == wave-relative LDS
// byte offset (ISA ch.10.2).
__device__ __forceinline__ uint32_t lds_offset_of(const void* p) {
    return (uint32_t)(uintptr_t)p;
}

// Tensor Data Mover: DMA a 1D run of `elems_remaining` dwords (clamped by the
// hardware to tile_dim0 = TILE_EDGES, zero-filling LDS past the tensor end)
// from global memory into LDS. Descriptor per CDNA5 ISA.
// Wave-level op (EXEC ignored), TENSORcnt-tracked; issue from one wave only.
__device__ __forceinline__ void tdm_load_dwords_to_lds(uint32_t lds_off, const void* gaddr,
                                                       uint32_t elems_remaining) {
    typedef uint32_t u32x4 __attribute__((ext_vector_type(4)));
    typedef uint32_t u32x8 __attribute__((ext_vector_type(8)));
    const uint64_t ga = (uint64_t)(uintptr_t)gaddr;

    u32x4 g0;
    g0[0] = 1u;                                      // [1:0] count=1 (valid), no gather
    g0[1] = lds_off;                                 // [63:32]  lds_addr (bytes)
    g0[2] = (uint32_t)ga;                            // [95:64]  global_addr[31:0]
    g0[3] = (uint32_t)((ga >> 32) & 0x01FFFFFFu)     // [120:96] global_addr[56:32]
          | (2u << 30);                              // [127:126] type = 2 ("image")

    u32x8 g1;
    g1[0] = (2u << 16);                              // data_size=2 (4B); mask/flags 0
    g1[1] = (elems_remaining & 0xFFFFu) << 16;       // tensor_dim0[15:0]  @ bits 63:48
    g1[2] = (elems_remaining >> 16)                  // tensor_dim0[31:16] @ bits 79:64
          | (1u << 16);                              // tensor_dim1 = 1    @ bits 95:80
    g1[3] = ((uint32_t)TILE_EDGES << 16);            // tile_dim0 = 1024   @ bits 127:112
    g1[4] = 0u;                                      // tile_dim1/2 = 0 (unused)
    g1[5] = (uint32_t)TILE_EDGES;                    // tensor_dim0_stride
    g1[6] = 0u;
    g1[7] = 0u;

    asm volatile("tensor_load_to_lds %0, %1" :: "s"(g0), "s"(g1) : "memory");
}

// ---------------------------------------------------------------------------
// Kernels
// ---------------------------------------------------------------------------

// Degree accumulation: one thread per edge, fp32 atomics into zeroed buffers.
__global__ __launch_bounds__(256) void degree_kernel(
    const int* __restrict__ src, const int* __restrict__ dst,
    float* __restrict__ out_deg, float* __restrict__ in_deg, int E)
{
    int e = blockIdx.x * blockDim.x + threadIdx.x;
    if (e < E) {
        const int s = src[e];          // both index loads issued before the
        const int d = dst[e];          // asm atomics (overlap the two gathers)
        atomic_add_f32_dev(&out_deg[s], 1.0f);
        atomic_add_f32_dev(&in_deg[d], 1.0f);
    }
}

// In-place: deg -> max(deg,1)^-0.5
__global__ __launch_bounds__(256) void norm_kernel(
    float* __restrict__ src_norm, float* __restrict__ dst_norm, int N)
{
    int i = blockIdx.x * blockDim.x + threadIdx.x;
    if (i < N) {
        src_norm[i] = 1.0f / sqrtf(fmaxf(src_norm[i], 1.0f));
        dst_norm[i] = 1.0f / sqrtf(fmaxf(dst_norm[i], 1.0f));
    }
}

// out = w0 * feat ; hs = feat * src_norm   (float4-granular, 16 vec4 per node)
__global__ __launch_bounds__(256) void init_kernel(
    const float* __restrict__ feat, const float* __restrict__ src_norm,
    float* __restrict__ out, float* __restrict__ hs, float w0, int nv4)
{
    int i = blockIdx.x * blockDim.x + threadIdx.x;
    if (i < nv4) {
        int node = i >> 4;                       // HIDDEN/4 = 16 float4 per node
        float sn = src_norm[node];
        float4 f = ((const float4*)feat)[i];
        ((float4*)out)[i] = make_float4(w0 * f.x, w0 * f.y, w0 * f.z, w0 * f.w);
        ((float4*)hs)[i]  = make_float4(sn * f.x, sn * f.y, sn * f.z, sn * f.w);
    }
}

// Edge scatter: hn[dst] += hs[src].
// Block = 256 threads = 8 waves, owning a TILE_EDGES-edge tile.
//  - src index tile: staged by the Tensor Data Mover (1 DMA instr from wave 0,
//    hardware zero-fills past E -> no ragged-tail code).
//  - dst index tile: staged by per-lane global_load_async_to_lds_b128.
// Compute: 2 edges per wave; lanes 0-15 own edge e, lanes 16-31 own edge e+1;
// each lane moves one float4 of the 64-wide row (16 lanes x 16B = 256B row).
// Gather = one coalesced b128 load/lane; scatter = four fp32 L2 atomics/lane.
__global__ __launch_bounds__(256) void edge_scatter(
    const int* __restrict__ src, const int* __restrict__ dst,
    const float* __restrict__ hs, float* __restrict__ hn, int E)
{
    __shared__ int s_src[TILE_EDGES];
    __shared__ int s_dst[TILE_EDGES];

    const int t    = threadIdx.x;
    const int tile = blockIdx.x * TILE_EDGES;
    const int nloc = min(TILE_EDGES, E - tile);

    // --- TDM DMA: src index tile (wave 0 only; EXEC-independent wave op) ---
    if (t < 32) {
        tdm_load_dwords_to_lds(lds_offset_of(&s_src[0]), src + tile, (uint32_t)(E - tile));
    }

    // --- async-to-LDS: dst index tile, 16B per lane --------------------------
    {
        const int base = tile + t * 4;
        if (base + 4 <= E) {
            async_load_b128_to_lds(lds_offset_of(&s_dst[t * 4]), dst + base);
        } else {
            for (int k = 0; k < 4; ++k) {
                int idx = base + k;
                if (idx < E) s_dst[t * 4 + k] = dst[idx];
            }
        }
    }

    wait_asynccnt0();           // every wave drains its own async copies
    if (t < 32) wait_tensorcnt0();
    __syncthreads();            // all staging visible to all 8 waves

    const int lane = t & 31;
    const int wid  = t >> 5;
    const int sub  = lane >> 4;          // which edge of the wave's pair
    const int foff = (lane & 15) * 4;    // float4 slot within the 64-wide row

    for (int e = wid * 2; e < nloc; e += 16) {
        const int my = e + sub;
        if (my < nloc) {
            const int s = s_src[my];     // LDS broadcast per half-wave
            const int d = s_dst[my];
            if (my + 16 < nloc) {        // hide random-gather latency
                const int sn = s_src[my + 16];
                __builtin_prefetch(hs + (size_t)sn * HIDDEN + foff, 0, 3);
            }
            const float4 v = *(const float4*)(hs + (size_t)s * HIDDEN + foff);
            float* p = hn + (size_t)d * HIDDEN + foff;
            atomic_add_f32_dev(p,     v.x);
            atomic_add_f32_dev(p + 1, v.y);
            atomic_add_f32_dev(p + 2, v.z);
            atomic_add_f32_dev(p + 3, v.w);
        }
    }
}

// h = hn * dst_norm ; out += w * h ; hs = h * src_norm ; hn = 0 (for next step)
__global__ __launch_bounds__(256) void node_update(
    float* __restrict__ hn, const float* __restrict__ dst_norm,
    const float* __restrict__ src_norm, float* __restrict__ out,
    float* __restrict__ hs, float w, int nv4)
{
    int i = blockIdx.x * blockDim.x + threadIdx.x;
    if (i < nv4) {
        int node = i >> 4;
        float dn = dst_norm[node];
        float sn = src_norm[node];
        float4 v = ((float4*)hn)[i];
        ((float4*)hn)[i] = make_float4(0.f, 0.f, 0.f, 0.f);   // fused re-zero
        v.x *= dn; v.y *= dn; v.z *= dn; v.w *= dn;
        float4 o = ((float4*)out)[i];
        o.x += w * v.x; o.y += w * v.y; o.z += w * v.z; o.w += w * v.w;
        ((float4*)out)[i] = o;
        ((float4*)hs)[i] = make_float4(sn * v.x, sn * v.y, sn * v.z, sn * v.w);
    }
}

// ---------------------------------------------------------------------------
// Launcher
// ---------------------------------------------------------------------------
extern "C" void kernel_launch(void* const* d_in, const int* in_sizes, int n_in,
                              void* d_out, int out_size, void* d_ws, size_t ws_size,
                              hipStream_t stream)
{
    (void)n_in; (void)out_size; (void)ws_size;
    const float* feat = (const float*)d_in[0];
    const int*   src  = (const int*)d_in[1];
    const int*   dst  = (const int*)d_in[2];
    float*       out  = (float*)d_out;

    const int N = in_sizes[0] / HIDDEN;
    const int E = in_sizes[1];

    // workspace layout: [src_norm N][dst_norm N][hs N*64][hn N*64]  (~52 MB)
    float* src_norm = (float*)d_ws;
    float* dst_norm = src_norm + N;
    float* hs       = dst_norm + N;
    float* hn       = hs + (size_t)N * HIDDEN;

    // host-side log weights (deterministic constants)
    double logs[KSTEPS + 1], denom = 0.0;
    for (int i = 0; i <= KSTEPS; ++i) { logs[i] = log(BETA_W + (double)(i + 1)); denom += logs[i]; }
    float w[KSTEPS + 1];
    for (int i = 0; i <= KSTEPS; ++i) w[i] = (float)(logs[i] / denom);

    // zero degree buffers + scatter accumulator (graph-capture-safe stream ops)
    hipMemsetAsync(src_norm, 0, 2 * (size_t)N * sizeof(float), stream);
    hipMemsetAsync(hn, 0, (size_t)N * HIDDEN * sizeof(float), stream);

    degree_kernel<<<(E + 255) / 256, 256, 0, stream>>>(src, dst, src_norm, dst_norm, E);
    norm_kernel<<<(N + 255) / 256, 256, 0, stream>>>(src_norm, dst_norm, N);

    const int nv4 = N * (HIDDEN / 4);
    init_kernel<<<(nv4 + 255) / 256, 256, 0, stream>>>(feat, src_norm, out, hs, w[0], nv4);

    const int edge_blocks = (E + TILE_EDGES - 1) / TILE_EDGES;
    for (int step = 0; step < KSTEPS; ++step) {
        edge_scatter<<<edge_blocks, 256, 0, stream>>>(src, dst, hs, hn, E);
        node_update<<<(nv4 + 255) / 256, 256, 0, stream>>>(hn, dst_norm, src_norm, out,
                                                           hs, w[step + 1], nv4);
    }
}